// GIN_43310450213482
// MI455X (gfx1250) — compile-verified
//
#include <hip/hip_runtime.h>
#include <math.h>

#define N_NODES   250000
#define N_EDGES   4000000
#define F_IN      128
#define H         16
#define N_GRAPHS  1024
#define N_TILES   (N_NODES / 16)   // 15625, exact
#define BN_EPS    1e-5f

typedef float v2f __attribute__((ext_vector_type(2)));
typedef float v8f __attribute__((ext_vector_type(8)));

// D = A(16x4 f32) * B(4x16 f32) + C(16x16 f32)
static __device__ __forceinline__ v8f wmma4(v2f a, v2f b, v8f c) {
  return __builtin_amdgcn_wmma_f32_16x16x4_f32(false, a, false, b, (short)0, c,
                                               false, false);
}

// ---- Stage A: y = x @ c1_w1   (250000x128 @ 128x16), one wave per 16 rows ----
__global__ __launch_bounds__(32) void k_in_gemm(const float* __restrict__ x,
                                                const float* __restrict__ w,
                                                float* __restrict__ y) {
  const int tile = blockIdx.x;
  const int lane = threadIdx.x;
  const int ln = lane & 15;        // A row / B,C column
  const int half = lane >> 4;
  const float* xr = x + (size_t)(tile * 16 + ln) * F_IN;
  v8f c = {};
  for (int k = 0; k < F_IN; k += 4) {
    const int kk = k + half * 2;
    v2f a; a.x = xr[kk]; a.y = xr[kk + 1];
    v2f b; b.x = w[kk * H + ln]; b.y = w[(kk + 1) * H + ln];
    c = wmma4(a, b, c);
  }
#pragma unroll
  for (int v = 0; v < 8; ++v) {
    const int m = v + half * 8;
    y[(size_t)(tile * 16 + m) * H + ln] = c[v];
  }
}

// ---- init agg = y (float4 copy) ----
__global__ void k_copy4(const float4* __restrict__ s, float4* __restrict__ d, int n) {
  int i = blockIdx.x * 256 + threadIdx.x;
  if (i < n) d[i] = s[i];
}

// ---- edge scatter: agg[dst] += y[src], one thread per (edge,feature) ----
__global__ __launch_bounds__(256) void k_edge(const int* __restrict__ src,
                                              const int* __restrict__ dst,
                                              const float* __restrict__ y,
                                              float* __restrict__ agg) {
  const int gid = blockIdx.x * 256 + threadIdx.x;   // exactly 64M threads
  const int e = gid >> 4;
  const int f = gid & 15;
  const int s = src[e];
  const int d = dst[e];
  atomicAdd(&agg[(size_t)d * H + f], y[(size_t)s * H + f]);
}

// ---- fused GIN node MLP + BatchNorm + pooled readout. One wave / 16 nodes ----
// layer0: w1 == nullptr  -> h1 = relu(agg + b1)        (w1 hoisted pre-agg)
// else:                     h1 = relu(agg @ w1 + b1)
// then h2 = relu(h1 @ w2 + b2);  out = (h2-mean)*gamma/sqrt(var+eps)+beta
__global__ __launch_bounds__(32) void k_mlp(const float* __restrict__ agg,
                                            float* __restrict__ yout,
                                            const float* __restrict__ w1,
                                            const float* __restrict__ b1,
                                            const float* __restrict__ w2,
                                            const float* __restrict__ b2,
                                            const float* __restrict__ gamma,
                                            const float* __restrict__ beta,
                                            const float* __restrict__ mean,
                                            const float* __restrict__ var,
                                            const int* __restrict__ batch,
                                            float* __restrict__ rbuf, int roff) {
  __shared__ float tile[16 * 17];
  const int t = blockIdx.x;
  const int lane = threadIdx.x;
  const int ln = lane & 15;
  const int half = lane >> 4;
  const float bias1 = b1[ln];

  if (w1) {
    const float* ar = agg + (size_t)(t * 16 + ln) * H;
    v8f c = {};
    for (int k = 0; k < H; k += 4) {
      const int kk = k + half * 2;
      v2f a; a.x = ar[kk]; a.y = ar[kk + 1];
      v2f b; b.x = w1[kk * H + ln]; b.y = w1[(kk + 1) * H + ln];
      c = wmma4(a, b, c);
    }
#pragma unroll
    for (int v = 0; v < 8; ++v) {
      const int m = v + half * 8;
      tile[m * 17 + ln] = fmaxf(c[v] + bias1, 0.f);
    }
  } else {
#pragma unroll
    for (int v = 0; v < 8; ++v) {
      const int m = v + half * 8;
      tile[m * 17 + ln] = fmaxf(agg[(size_t)(t * 16 + m) * H + ln] + bias1, 0.f);
    }
  }
  // in-wave LDS visibility: wait for the wave's DS stores to land
  __asm__ volatile("s_wait_dscnt 0" ::: "memory");

  v8f c2 = {};
  for (int k = 0; k < H; k += 4) {
    const int kk = k + half * 2;
    v2f a; a.x = tile[ln * 17 + kk]; a.y = tile[ln * 17 + kk + 1];
    v2f b; b.x = w2[kk * H + ln]; b.y = w2[(kk + 1) * H + ln];
    c2 = wmma4(a, b, c2);
  }

  const float sc = gamma[ln] * rsqrtf(var[ln] + BN_EPS);
  const float mn = mean[ln];
  const float bt = beta[ln];
  const float bias2 = b2[ln];
#pragma unroll
  for (int v = 0; v < 8; ++v) {
    const int m = v + half * 8;
    const int node = t * 16 + m;
    const float h = fmaxf(c2[v] + bias2, 0.f);
    const float o = (h - mn) * sc + bt;
    yout[(size_t)node * H + ln] = o;
    atomicAdd(&rbuf[(size_t)batch[node] * 80 + roff + ln], o);  // pooled readout
  }
}

// ---- zero fill ----
__global__ void k_zero(float* p, int n) {
  int i = blockIdx.x * 256 + threadIdx.x;
  if (i < n) p[i] = 0.f;
}

// ---- head: relu(r@W1+b1) @ W2 + b2 -> log_softmax. One block per graph ----
__global__ __launch_bounds__(128) void k_head(const float* __restrict__ r,
                                              const float* __restrict__ w1,
                                              const float* __restrict__ b1,
                                              const float* __restrict__ w2,
                                              const float* __restrict__ b2,
                                              float* __restrict__ out) {
  __shared__ float rr[80], h1[80], lg[10];
  __shared__ float lse;
  const int g = blockIdx.x;
  const int t = threadIdx.x;
  if (t < 80) rr[t] = r[(size_t)g * 80 + t];
  __syncthreads();
  if (t < 80) {
    float a = b1[t];
    for (int k = 0; k < 80; ++k) a += rr[k] * w1[k * 80 + t];
    h1[t] = fmaxf(a, 0.f);
  }
  __syncthreads();
  if (t < 10) {
    float a = b2[t];
    for (int k = 0; k < 80; ++k) a += h1[k] * w2[k * 10 + t];
    lg[t] = a;
  }
  __syncthreads();
  if (t == 0) {
    float m = lg[0];
    for (int i = 1; i < 10; ++i) m = fmaxf(m, lg[i]);
    float s = 0.f;
    for (int i = 0; i < 10; ++i) s += expf(lg[i] - m);
    lse = m + logf(s);
  }
  __syncthreads();
  if (t < 10) out[(size_t)g * 10 + t] = lg[t] - lse;
}

extern "C" void kernel_launch(void* const* d_in, const int* in_sizes, int n_in,
                              void* d_out, int out_size, void* d_ws, size_t ws_size,
                              hipStream_t stream) {
  const float* x      = (const float*)d_in[0];
  const int*   ei     = (const int*)d_in[1];
  const int*   batch  = (const int*)d_in[2];
  const float* c1_w1  = (const float*)d_in[3];
  const float* c1_b1  = (const float*)d_in[4];
  const float* c1_w2  = (const float*)d_in[5];
  const float* c1_b2  = (const float*)d_in[6];
  const float* c1_g   = (const float*)d_in[7];
  const float* c1_be  = (const float*)d_in[8];
  const float* c1_m   = (const float*)d_in[9];
  const float* c1_v   = (const float*)d_in[10];
  const float* cs_w1  = (const float*)d_in[11];
  const float* cs_b1  = (const float*)d_in[12];
  const float* cs_w2  = (const float*)d_in[13];
  const float* cs_b2  = (const float*)d_in[14];
  const float* cs_g   = (const float*)d_in[15];
  const float* cs_be  = (const float*)d_in[16];
  const float* cs_m   = (const float*)d_in[17];
  const float* cs_v   = (const float*)d_in[18];
  const float* lin1_w = (const float*)d_in[19];
  const float* lin1_b = (const float*)d_in[20];
  const float* lin2_w = (const float*)d_in[21];
  const float* lin2_b = (const float*)d_in[22];

  const int* src = ei;
  const int* dst = ei + N_EDGES;

  float* y    = (float*)d_ws;                    // 250000*16 f32 = 16 MB
  float* agg  = y + (size_t)N_NODES * H;         // 16 MB
  float* rbuf = agg + (size_t)N_NODES * H;       // 1024*80 f32

  // zero pooled-readout accumulator (ws is poisoned by harness)
  k_zero<<<(N_GRAPHS * 80 + 255) / 256, 256, 0, stream>>>(rbuf, N_GRAPHS * 80);

  // hoisted first-layer weight: y = x @ c1_w1
  k_in_gemm<<<N_TILES, 32, 0, stream>>>(x, c1_w1, y);

  const int nvec4 = N_NODES * H / 4;  // 1,000,000
  for (int l = 0; l < 5; ++l) {
    k_copy4<<<(nvec4 + 255) / 256, 256, 0, stream>>>((const float4*)y,
                                                     (float4*)agg, nvec4);
    k_edge<<<(N_EDGES * 16) / 256, 256, 0, stream>>>(src, dst, y, agg);

    const float *w1p, *b1p, *w2p, *b2p, *gp, *bep, *mp, *vp;
    if (l == 0) {
      w1p = nullptr; b1p = c1_b1; w2p = c1_w2; b2p = c1_b2;
      gp = c1_g; bep = c1_be; mp = c1_m; vp = c1_v;
    } else {
      const int j = l - 1;
      w1p = cs_w1 + j * H * H; b1p = cs_b1 + j * H;
      w2p = cs_w2 + j * H * H; b2p = cs_b2 + j * H;
      gp = cs_g + j * H; bep = cs_be + j * H;
      mp = cs_m + j * H; vp = cs_v + j * H;
    }
    k_mlp<<<N_TILES, 32, 0, stream>>>(agg, y, w1p, b1p, w2p, b2p,
                                      gp, bep, mp, vp, batch, rbuf, l * 16);
  }

  k_head<<<N_GRAPHS, 128, 0, stream>>>(rbuf, lin1_w, lin1_b, lin2_w, lin2_b,
                                       (float*)d_out);
}